// DetectionLoss_84404697301296
// MI455X (gfx1250) — compile-verified
//
#include <hip/hip_runtime.h>
#include <hip/hip_bf16.h>
#include <stdint.h>

#define GG 7
#define AAN 2
#define CCLS 3
#define MMAX 20
#define CELLS (GG * GG * AAN)   /* 98 */
#define PSTRIDE 8               /* 5 + C */
#define SLAB (CELLS * PSTRIDE)  /* 784 floats = 3136 B */
#define BLK 128

typedef __attribute__((ext_vector_type(2))) float v2f;
typedef __attribute__((ext_vector_type(8))) float v8f;

// Exact parameter type of the async-to-LDS builtins on this toolchain:
// pointer to <4 x i32>, AS1 (printed as __device__) / AS3 (__shared__).
typedef int v4i __attribute__((vector_size(4 * sizeof(int))));
typedef __attribute__((address_space(1))) v4i* gv4i_ptr;
typedef __attribute__((address_space(3))) v4i* lv4i_ptr;

__device__ __forceinline__ float softplusf(float x) {
    // numerically stable log(1+exp(x))
    return fmaxf(x, 0.0f) + log1pf(expf(-fabsf(x)));
}

__global__ void __launch_bounds__(64) det_zero_accum(float* acc) {
    if (threadIdx.x < 8) acc[threadIdx.x] = 0.0f;
}

__global__ void __launch_bounds__(BLK) det_loss_main(
    const float* __restrict__ preds,
    const float* __restrict__ tboxes,
    const int*   __restrict__ tlabels,
    const int*   __restrict__ nobjs,
    const float* __restrict__ anchors,
    const float* __restrict__ cw,
    float* __restrict__ acc)
{
    __shared__ float s_pred[SLAB];
    __shared__ float s_mask[CELLS];
    __shared__ float s_tb[CELLS][4];
    __shared__ int   s_cls[CELLS];
    __shared__ float s_acc[5];

    const int b   = blockIdx.x;
    const int tid = threadIdx.x;

    // ---- Stage this batch's prediction slab into LDS (async path if present)
    const float* gslab = preds + (size_t)b * SLAB;
#if __has_builtin(__builtin_amdgcn_global_load_async_to_lds_b128)
    for (int k = tid; k < SLAB / 4; k += BLK) {
        __builtin_amdgcn_global_load_async_to_lds_b128(
            (gv4i_ptr)(float*)(gslab + 4 * k),
            (lv4i_ptr)(s_pred + 4 * k),
            0, 0);
    }
#if __has_builtin(__builtin_amdgcn_s_wait_asynccnt)
    __builtin_amdgcn_s_wait_asynccnt(0);
#else
    asm volatile("s_wait_asynccnt 0" ::: "memory");
#endif
#else
    for (int k = tid; k < SLAB / 4; k += BLK) {
        ((float4*)s_pred)[k] = ((const float4*)gslab)[k];
    }
#endif

    // ---- Zero the assignment table (disjoint LDS region; runs after the async wait)
    if (tid < CELLS) {
        s_mask[tid] = 0.0f;
        s_cls[tid]  = 0;
        s_tb[tid][0] = 0.0f; s_tb[tid][1] = 0.0f;
        s_tb[tid][2] = 0.0f; s_tb[tid][3] = 0.0f;
    }
    if (tid < 5) s_acc[tid] = 0.0f;
    __syncthreads();

    // ---- Sequential target assignment (deterministic last-write-wins,
    //      matches the reference scatter's scan order). One thread, M<=20.
    if (tid == 0) {
        const float a0w = anchors[0], a0h = anchors[1];
        const float a1w = anchors[2], a1h = anchors[3];
        int n = nobjs[b];
        n = n < 0 ? 0 : (n > MMAX ? MMAX : n);
        const float* tb = tboxes  + (size_t)b * MMAX * 4;
        const int*   tl = tlabels + (size_t)b * MMAX;
        for (int m = 0; m < n; ++m) {
            float x1 = tb[m * 4 + 0], y1 = tb[m * 4 + 1];
            float x2 = tb[m * 4 + 2], y2 = tb[m * 4 + 3];
            float w = x2 - x1, h = y2 - y1;
            if (!(w > 0.0f && h > 0.0f)) continue;
            float cx = 0.5f * (x1 + x2), cy = 0.5f * (y1 + y2);
            int gi = (int)floorf(cy * (float)GG);
            int gj = (int)floorf(cx * (float)GG);
            gi = gi < 0 ? 0 : (gi > GG - 1 ? GG - 1 : gi);
            gj = gj < 0 ? 0 : (gj > GG - 1 ? GG - 1 : gj);
            float wg = w * (float)GG, hg = h * (float)GG;
            float i0 = fminf(wg, a0w) * fminf(hg, a0h);
            float u0 = wg * hg + a0w * a0h - i0;
            float i1 = fminf(wg, a1w) * fminf(hg, a1h);
            float u1 = wg * hg + a1w * a1h - i1;
            // argmax with first-max tie-break (strict >)
            int best = (i1 / (u1 + 1e-6f)) > (i0 / (u0 + 1e-6f)) ? 1 : 0;
            float aw = best ? a1w : a0w;
            float ah = best ? a1h : a0h;
            int cell = (gi * GG + gj) * AAN + best;
            s_mask[cell]  = 1.0f;
            s_tb[cell][0] = cx * (float)GG - (float)gj;
            s_tb[cell][1] = cy * (float)GG - (float)gi;
            s_tb[cell][2] = logf(fmaxf(wg, 0.01f) / (aw + 1e-6f));
            s_tb[cell][3] = logf(fmaxf(hg, 0.01f) / (ah + 1e-6f));
            s_cls[cell]   = tl[m];
        }
    }
    __syncthreads();

    // ---- Per-cell losses from LDS
    float l_bbox = 0.0f, l_obj = 0.0f, l_noobj = 0.0f, l_cls = 0.0f, npos = 0.0f;
    for (int c = tid; c < CELLS; c += BLK) {
        const float* p = &s_pred[c * PSTRIDE];
        float mobj = s_mask[c];
        float po = p[0];
        l_obj   += mobj * softplusf(-po);
        l_noobj += (1.0f - mobj) * softplusf(po);
        if (mobj > 0.0f) {
#pragma unroll
            for (int j = 0; j < 4; ++j) {
                float d  = p[1 + j] - s_tb[c][j];
                float ad = fabsf(d);
                l_bbox += (ad < 1.0f) ? 0.5f * d * d : ad - 0.5f;
            }
            float c0 = p[5], c1 = p[6], c2 = p[7];
            float mx  = fmaxf(c0, fmaxf(c1, c2));
            float lse = mx + logf(expf(c0 - mx) + expf(c1 - mx) + expf(c2 - mx));
            int t = s_cls[c];
            l_cls += cw[t] * (lse - p[5 + t]);
            npos  += 1.0f;
        }
    }
    // control flow is converged here (loop exits uniformly) -> EXEC all-ones

    // ---- Wave-level reduction of the 5 partials via WMMA against ones.
    // A(16x4) holds 2 values/lane; B = ones -> D[m][n] = rowsum(A,m).
    // lanes 0..15 see D[0..7][*] in d[0..7], lanes 16..31 see D[8..15][*].
    float comps[5] = { l_bbox, l_obj, l_noobj, l_cls, npos };
    float wtot[5];
#if __has_builtin(__builtin_amdgcn_wmma_f32_16x16x4_f32)
    v2f ones; ones.x = 1.0f; ones.y = 1.0f;
#pragma unroll
    for (int i = 0; i < 5; ++i) {
        v2f a; a.x = comps[i]; a.y = 0.0f;
        v8f cz = {};
        v8f d = __builtin_amdgcn_wmma_f32_16x16x4_f32(
            false, a, false, ones, (short)0, cz, false, false);
        float s = d[0] + d[1] + d[2] + d[3] + d[4] + d[5] + d[6] + d[7];
        wtot[i] = s + __shfl_xor(s, 16, 32);
    }
#else
#pragma unroll
    for (int i = 0; i < 5; ++i) {
        float s = comps[i];
        for (int off = 16; off >= 1; off >>= 1) s += __shfl_xor(s, off, 32);
        wtot[i] = s;
    }
#endif

    if ((tid & 31) == 0) {
#pragma unroll
        for (int i = 0; i < 5; ++i) atomicAdd(&s_acc[i], wtot[i]);
    }
    __syncthreads();
    if (tid < 5) atomicAdd(&acc[tid], s_acc[tid]);
}

__global__ void __launch_bounds__(64) det_loss_final(const float* __restrict__ acc,
                                                     float* __restrict__ out) {
    if (threadIdx.x == 0 && blockIdx.x == 0) {
        float np = fmaxf(acc[4], 1.0f);
        out[0] = (5.0f * acc[0] + 1.0f * acc[1] + 0.5f * acc[2] + 2.0f * acc[3]) / np;
    }
}

extern "C" void kernel_launch(void* const* d_in, const int* in_sizes, int n_in,
                              void* d_out, int out_size, void* d_ws, size_t ws_size,
                              hipStream_t stream) {
    const float* preds  = (const float*)d_in[0];
    const float* tboxes = (const float*)d_in[1];
    const int*   tlab   = (const int*)d_in[2];
    const int*   nobj   = (const int*)d_in[3];
    const float* anch   = (const float*)d_in[4];
    const float* cwts   = (const float*)d_in[5];
    float* acc = (float*)d_ws;

    const int B = in_sizes[3];  // num_objs is (B,)

    det_zero_accum<<<1, 64, 0, stream>>>(acc);
    det_loss_main<<<B, BLK, 0, stream>>>(preds, tboxes, tlab, nobj, anch, cwts, acc);
    det_loss_final<<<1, 64, 0, stream>>>(acc, (float*)d_out);
}